// Decoder_41266045780768
// MI455X (gfx1250) — compile-verified
//
#include <hip/hip_runtime.h>
#include <math.h>

typedef __attribute__((ext_vector_type(2))) float v2f;
typedef __attribute__((ext_vector_type(8))) float v8f;

#define HID   400
#define BATCH 256
#define NGATE 1600          // 4*HID
#define VOCAB 32000

// ---------------------------------------------------------------------------
// Kernel 1: pred = log_softmax over a singleton axis == exactly zero.
// Pure bandwidth: 32.8 MB of zeros. float4 grid-stride fill.
// ---------------------------------------------------------------------------
__global__ __launch_bounds__(256) void zero_pred_kernel(float4* __restrict__ out, int n4) {
    int i      = blockIdx.x * blockDim.x + threadIdx.x;
    int stride = gridDim.x * blockDim.x;
    float4 z = make_float4(0.f, 0.f, 0.f, 0.f);
    for (; i < n4; i += stride) out[i] = z;
}

// ---------------------------------------------------------------------------
// Kernel 2: gates[b,n] = sum_k emb[prev[b]][k]*w_ih[n][k]
//                      + sum_k h0[b][k]     *w_hh[n][k]  + b_ih[n] + b_hh[n]
// One wave32 per 16x16 tile of gates, V_WMMA_F32_16X16X4_F32, K-loop of 4.
// A 16x4 f32 fragment: lane<16 -> {A[m][k0],A[m][k0+1]}, lane>=16 -> k0+2,k0+3
// B 4x16 f32 fragment: same k-pattern, n = lane&15. W is [N,K] row-major, so
// B[k][n] = W[n][k] -> contiguous float2 per lane for both A and B.
// C/D layout: VGPR r holds M = r + 8*(lane>>4), N = lane&15.
// ---------------------------------------------------------------------------
__global__ __launch_bounds__(128) void lstm_gates_wmma(
    const int*   __restrict__ prev,
    const float* __restrict__ emb,
    const float* __restrict__ h0,
    const float* __restrict__ w_ih,
    const float* __restrict__ w_hh,
    const float* __restrict__ b_ih,
    const float* __restrict__ b_hh,
    float*       __restrict__ gates)
{
    const int lane  = threadIdx.x & 31;
    const int wave  = threadIdx.x >> 5;
    const int mbase = blockIdx.y * 16;                 // batch tile
    const int nbase = blockIdx.x * 64 + wave * 16;     // gate-column tile

    const int l16  = lane & 15;
    const int koff = (lane >> 4) * 2;                  // 0 for lanes 0-15, 2 for 16-31

    const int    rowA   = mbase + l16;                 // batch row for A fragment
    const int    embRow = prev[rowA];                  // embedding gather
    const float* pX = emb + (long)embRow * HID;
    const float* pH = h0  + (long)rowA   * HID;

    const int    ncol = nbase + l16;                   // gate column for B fragment
    const float* pW1  = w_ih + (long)ncol * HID;
    const float* pW2  = w_hh + (long)ncol * HID;

    v8f acc = {};
#pragma unroll 4
    for (int kb = 0; kb < HID; kb += 4) {
        v2f a1 = *(const v2f*)(pX  + kb + koff);
        v2f b1 = *(const v2f*)(pW1 + kb + koff);
        acc = __builtin_amdgcn_wmma_f32_16x16x4_f32(
            false, a1, false, b1, (short)0, acc, false, false);
        v2f a2 = *(const v2f*)(pH  + kb + koff);
        v2f b2 = *(const v2f*)(pW2 + kb + koff);
        acc = __builtin_amdgcn_wmma_f32_16x16x4_f32(
            false, a2, false, b2, (short)0, acc, false, false);
    }

    const float bias = b_ih[ncol] + b_hh[ncol];
    const int   mtop = mbase + (lane >> 4) * 8;
#pragma unroll
    for (int r = 0; r < 8; ++r)
        gates[(long)(mtop + r) * NGATE + ncol] = acc[r] + bias;
}

// ---------------------------------------------------------------------------
// Kernel 3: LSTM cell nonlinearity (PyTorch gate order i,f,g,o).
// ---------------------------------------------------------------------------
__global__ __launch_bounds__(256) void lstm_cell_ew(
    const float* __restrict__ gates,
    const float* __restrict__ c0,
    float*       __restrict__ h_out,
    float*       __restrict__ c_out)
{
    int t = blockIdx.x * blockDim.x + threadIdx.x;
    if (t >= BATCH * HID) return;
    int b = t / HID;
    int j = t - b * HID;
    const float* g = gates + (long)b * NGATE;
    float ig = 1.f / (1.f + expf(-g[j]));
    float fg = 1.f / (1.f + expf(-g[HID + j]));
    float gg = tanhf(g[2 * HID + j]);
    float og = 1.f / (1.f + expf(-g[3 * HID + j]));
    float c  = fg * c0[t] + ig * gg;
    float h  = og * tanhf(c);
    h_out[t] = h;
    c_out[t] = c;
}

// ---------------------------------------------------------------------------
extern "C" void kernel_launch(void* const* d_in, const int* in_sizes, int n_in,
                              void* d_out, int out_size, void* d_ws, size_t ws_size,
                              hipStream_t stream) {
    const int*   prev = (const int*)  d_in[0];
    const float* h0   = (const float*)d_in[1];
    const float* c0   = (const float*)d_in[2];
    // d_in[3] encoder_outputs: dead (softmax over singleton axis -> alpha==1,
    // and pred is log_softmax over singleton axis -> exactly 0 regardless)
    const float* emb  = (const float*)d_in[4];
    const float* w_ih = (const float*)d_in[5];
    const float* w_hh = (const float*)d_in[6];
    const float* b_ih = (const float*)d_in[7];
    const float* b_hh = (const float*)d_in[8];
    // d_in[9..14] (W1, W2, W) are dead for all three outputs.

    float* out   = (float*)d_out;
    float* gates = (float*)d_ws;                       // 256*1600*4 = 1.6 MB

    // pred region: BATCH * 1 * VOCAB zeros
    const int n4 = (BATCH * VOCAB) / 4;                // 2,048,000 float4s
    zero_pred_kernel<<<2048, 256, 0, stream>>>((float4*)out, n4);

    // gates GEMM via WMMA: grid = (1600/64 ntiles, 256/16 mtiles), 4 waves/block
    dim3 grid(NGATE / 64, BATCH / 16);
    lstm_gates_wmma<<<grid, 128, 0, stream>>>(prev, emb, h0, w_ih, w_hh,
                                              b_ih, b_hh, gates);

    // h then c appended after pred, each [1, B, H]
    float* h_out = out + (long)BATCH * VOCAB;
    float* c_out = h_out + BATCH * HID;
    lstm_cell_ew<<<(BATCH * HID + 255) / 256, 256, 0, stream>>>(gates, c0,
                                                                h_out, c_out);
}